// HANLayer_41291815583854
// MI455X (gfx1250) — compile-verified
//
#include <hip/hip_runtime.h>
#include <math.h>

// HAN layer: P=3 metapaths, N nodes, E edges/path, H=4 heads, D=64, IN=256.
#define HEADS 4
#define INCH  256
#define PDIM  3

typedef __attribute__((ext_vector_type(2))) float v2f;
typedef __attribute__((ext_vector_type(8))) float v8f;

// ---------------------------------------------------------------- utilities
__global__ void fill_f32(float* __restrict__ p, float v, size_t n) {
    size_t i = (size_t)blockIdx.x * blockDim.x + threadIdx.x;
    if (i < n) p[i] = v;
}

__device__ __forceinline__ float atomicMaxF(float* addr, float v) {
    // monotonic int/uint mapping trick (works for mixed signs, init = -inf)
    if (v >= 0.0f)
        return __int_as_float(atomicMax((int*)addr, __float_as_int(v)));
    else
        return __uint_as_float(atomicMin((unsigned int*)addr, __float_as_uint(v)));
}

// ---------------------------------------------------------------- GEMM (WMMA)
// feat[M,256] = h[M,256] @ W[256,256] using v_wmma_f32_16x16x4_f32.
// One wave computes a 16(M) x 64(N) strip: 4 accumulators reuse each A fragment
// 4x (1 v2f A load + 8 scalar B loads feed 4 WMMAs per k-step).
__global__ void gemm_feat_wmma(const float* __restrict__ A,
                               const float* __restrict__ B,
                               float* __restrict__ C, int Mtiles) {
    const int wave = threadIdx.x >> 5;
    const int lane = threadIdx.x & 31;
    const int mt = blockIdx.y * 8 + wave;
    if (mt >= Mtiles) return;                 // wave-uniform exit: EXEC all-1s for WMMA
    const int m0   = mt * 16;
    const int n0   = blockIdx.x * 64;
    const int col  = lane & 15;               // A row index / B,D column index
    const int half = lane >> 4;

    const float* Arow = A + (size_t)(m0 + col) * INCH;
    v8f acc0 = {}, acc1 = {}, acc2 = {}, acc3 = {};

    #pragma unroll 2
    for (int k = 0; k < INCH; k += 4) {
        // A 16x4 f32: lanes 0-15 hold K=k,k+1 (v0,v1); lanes 16-31 hold K=k+2,k+3
        v2f a = *(const v2f*)(Arow + k + 2 * half);
        // B 4x16 f32: v0 = row K=k (half 0) / k+2 (half 1); v1 = k+1 / k+3
        const float* Brow0 = B + (size_t)(k + 2 * half) * INCH + n0 + col;
        const float* Brow1 = Brow0 + INCH;
        v2f b0, b1, b2, b3;
        b0.x = Brow0[0];  b0.y = Brow1[0];
        b1.x = Brow0[16]; b1.y = Brow1[16];
        b2.x = Brow0[32]; b2.y = Brow1[32];
        b3.x = Brow0[48]; b3.y = Brow1[48];
        acc0 = __builtin_amdgcn_wmma_f32_16x16x4_f32(false, a, false, b0,
                                                     (short)0, acc0, false, false);
        acc1 = __builtin_amdgcn_wmma_f32_16x16x4_f32(false, a, false, b1,
                                                     (short)0, acc1, false, false);
        acc2 = __builtin_amdgcn_wmma_f32_16x16x4_f32(false, a, false, b2,
                                                     (short)0, acc2, false, false);
        acc3 = __builtin_amdgcn_wmma_f32_16x16x4_f32(false, a, false, b3,
                                                     (short)0, acc3, false, false);
    }

    // D 16x16 f32: VGPR i -> M = i (lanes 0-15) / 8+i (lanes 16-31), N = lane&15
    #pragma unroll
    for (int i = 0; i < 8; ++i) {
        float* Crow = C + (size_t)(m0 + i + half * 8) * INCH + n0 + col;
        Crow[0]  = acc0[i];
        Crow[16] = acc1[i];
        Crow[32] = acc2[i];
        Crow[48] = acc3[i];
    }
}

// ---------------------------------------------------------------- attention
__global__ void elr_kernel(const float* __restrict__ feat,
                           const float* __restrict__ al,
                           const float* __restrict__ ar,
                           float* __restrict__ el, float* __restrict__ er, int N) {
    int idx = blockIdx.x * blockDim.x + threadIdx.x;
    if (idx >= N * HEADS) return;
    int n = idx >> 2, hh = idx & 3;
    const float* f = feat + (size_t)n * INCH + hh * 64;
    float sl = 0.f, sr = 0.f;
    #pragma unroll 8
    for (int d = 0; d < 64; ++d) {
        float v = f[d];
        sl += v * al[hh * 64 + d];
        sr += v * ar[hh * 64 + d];
    }
    el[idx] = sl;
    er[idx] = sr;
}

__global__ void edge_max_kernel(const int* __restrict__ src, const int* __restrict__ dst,
                                const float* __restrict__ el, const float* __restrict__ er,
                                float* __restrict__ mmax, int E) {
    int e = blockIdx.x * blockDim.x + threadIdx.x;
    if (e >= E) return;
    int s = src[e], d = dst[e];
    #pragma unroll
    for (int hh = 0; hh < HEADS; ++hh) {
        float v = el[s * HEADS + hh] + er[d * HEADS + hh];
        v = (v > 0.f) ? v : 0.2f * v;          // leaky_relu(0.2)
        atomicMaxF(&mmax[d * HEADS + hh], v);
    }
}

__global__ void edge_expsum_kernel(const int* __restrict__ src, const int* __restrict__ dst,
                                   const float* __restrict__ el, const float* __restrict__ er,
                                   const float* __restrict__ mmax,
                                   float* __restrict__ denom, float* __restrict__ exb, int E) {
    int e = blockIdx.x * blockDim.x + threadIdx.x;
    if (e >= E) return;
    int s = src[e], d = dst[e];
    #pragma unroll
    for (int hh = 0; hh < HEADS; ++hh) {
        float v = el[s * HEADS + hh] + er[d * HEADS + hh];
        v = (v > 0.f) ? v : 0.2f * v;
        float ex = __expf(v - mmax[d * HEADS + hh]);
        exb[(size_t)e * HEADS + hh] = ex;
        atomicAdd(&denom[d * HEADS + hh], ex);
    }
}

// one block (256 threads) per edge: gather feat[src], scatter-add into agg[dst]
__global__ void edge_aggr_kernel(const int* __restrict__ src, const int* __restrict__ dst,
                                 const float* __restrict__ feat,
                                 const float* __restrict__ exb,
                                 const float* __restrict__ denom,
                                 float* __restrict__ agg, float* __restrict__ aout, int E) {
    int e = blockIdx.x;
    int c = threadIdx.x;
    int s = src[e], d = dst[e];
    int hh = c >> 6;
    float alpha = exb[(size_t)e * HEADS + hh] / denom[d * HEADS + hh];
    atomicAdd(&agg[(size_t)d * INCH + c], alpha * feat[(size_t)s * INCH + c]);
    if (c == 0) {
        float m = 0.f;
        #pragma unroll
        for (int k = 0; k < HEADS; ++k)
            m += exb[(size_t)e * HEADS + k] / denom[d * HEADS + k];
        aout[e] = 0.25f * m;                   // mean over heads
    }
}

// ---------------------------------------------------------------- batchnorm
__global__ void bn_stats_kernel(const float* __restrict__ agg, const float* __restrict__ h,
                                float* __restrict__ stats, int N) {
    int c = threadIdx.x;                       // 256 channels
    float s = 0.f, s2 = 0.f;
    for (int n = blockIdx.x; n < N; n += gridDim.x) {
        float y = agg[(size_t)n * INCH + c];
        y = (y > 0.f) ? y : 0.f;               // relu before residual
        y += h[(size_t)n * INCH + c];
        s += y;
        s2 += y * y;
    }
    atomicAdd(&stats[c], s);
    atomicAdd(&stats[INCH + c], s2);
}

__global__ void finalize_kernel(const float* __restrict__ agg, const float* __restrict__ h,
                                const float* __restrict__ stats,
                                const float* __restrict__ gamma, const float* __restrict__ beta,
                                const float* __restrict__ path_mask,
                                float* __restrict__ pooled, float* __restrict__ wout,
                                int N, int p) {
    int idx = blockIdx.x * blockDim.x + threadIdx.x;
    if (idx >= N * INCH) return;
    int n = idx >> 8, c = idx & 255;
    float invN = 1.0f / (float)N;
    float mu  = stats[c] * invN;
    float var = stats[INCH + c] * invN - mu * mu;
    float y = agg[idx];
    y = (y > 0.f) ? y : 0.f;
    y += h[idx];
    float xn = gamma[c] * (y - mu) * rsqrtf(var + 1e-5f) + beta[c];
    float pm0 = path_mask[n * PDIM + 0];
    float pm1 = path_mask[n * PDIM + 1];
    float pm2 = path_mask[n * PDIM + 2];
    float inv_rs = 1.0f / (pm0 + pm1 + pm2);
    float wnp = path_mask[n * PDIM + p] * inv_rs;
    if (p == 0) {
        pooled[idx] = wnp * xn;                // overwrite poison
        if (c < PDIM) wout[n * PDIM + c] = path_mask[n * PDIM + c] * inv_rs;
    } else {
        pooled[idx] += wnp * xn;               // ordered: same stream
    }
}

// ---------------------------------------------------------------- host
extern "C" void kernel_launch(void* const* d_in, const int* in_sizes, int n_in,
                              void* d_out, int out_size, void* d_ws, size_t ws_size,
                              hipStream_t stream) {
    const float* h        = (const float*)d_in[0];
    const float* path_msk = (const float*)d_in[1];
    const int*   src      = (const int*)  d_in[2];
    const int*   dst      = (const int*)  d_in[3];
    const float* fc_w     = (const float*)d_in[4];
    const float* attn_l   = (const float*)d_in[5];
    const float* attn_r   = (const float*)d_in[6];
    const float* bn_gamma = (const float*)d_in[7];
    const float* bn_beta  = (const float*)d_in[8];

    const int N = in_sizes[0] / INCH;          // 50000
    const int E = in_sizes[2] / PDIM;          // 800000

    float* pooled   = (float*)d_out;                       // [N,256]
    float* wout     = pooled + (size_t)N * INCH;           // [N,3]
    float* attn_out = wout + (size_t)N * PDIM;             // [3,E]

    float* ws    = (float*)d_ws;
    float* feat  = ws;                                     // N*256
    float* agg   = feat  + (size_t)N * INCH;               // N*256
    float* el    = agg   + (size_t)N * INCH;               // N*4
    float* er    = el    + (size_t)N * HEADS;              // N*4
    float* mmax  = er    + (size_t)N * HEADS;              // N*4
    float* denom = mmax  + (size_t)N * HEADS;              // N*4
    float* exb   = denom + (size_t)N * HEADS;              // E*4
    float* stats = exb   + (size_t)E * HEADS;              // 512

    const int Mtiles = (N + 15) / 16;
    dim3 gemm_grid(INCH / 64, (Mtiles + 7) / 8);           // (4, ceil(3125/8))

    const size_t nh = (size_t)N * HEADS;
    const size_t nc = (size_t)N * INCH;

    for (int p = 0; p < PDIM; ++p) {
        const float* Wp  = fc_w  + (size_t)p * INCH * INCH;
        const float* alp = attn_l + (size_t)p * HEADS * 64;
        const float* arp = attn_r + (size_t)p * HEADS * 64;
        const int*   sp  = src + (size_t)p * E;
        const int*   dp  = dst + (size_t)p * E;

        fill_f32<<<(int)((nh + 255) / 256), 256, 0, stream>>>(mmax, -INFINITY, nh);
        fill_f32<<<(int)((nh + 255) / 256), 256, 0, stream>>>(denom, 0.f, nh);
        fill_f32<<<(int)((nc + 255) / 256), 256, 0, stream>>>(agg, 0.f, nc);
        fill_f32<<<2, 256, 0, stream>>>(stats, 0.f, (size_t)2 * INCH);

        gemm_feat_wmma<<<gemm_grid, 256, 0, stream>>>(h, Wp, feat, Mtiles);
        elr_kernel<<<(N * HEADS + 255) / 256, 256, 0, stream>>>(feat, alp, arp, el, er, N);
        edge_max_kernel<<<(E + 255) / 256, 256, 0, stream>>>(sp, dp, el, er, mmax, E);
        edge_expsum_kernel<<<(E + 255) / 256, 256, 0, stream>>>(sp, dp, el, er, mmax,
                                                                denom, exb, E);
        edge_aggr_kernel<<<E, 256, 0, stream>>>(sp, dp, feat, exb, denom, agg,
                                                attn_out + (size_t)p * E, E);
        bn_stats_kernel<<<256, 256, 0, stream>>>(agg, h, stats, N);
        finalize_kernel<<<(int)((nc + 255) / 256), 256, 0, stream>>>(
            agg, h, stats, bn_gamma, bn_beta, path_msk, pooled, wout, N, p);
    }
}